// CNFLayer_17119739641883
// MI455X (gfx1250) — compile-verified
//
#include <hip/hip_runtime.h>
#include <hip/hip_bf16.h>

typedef float v2f __attribute__((ext_vector_type(2)));
typedef float v8f __attribute__((ext_vector_type(8)));

#define D 64

// ---------------------------------------------------------------------------
// Zero-fill (graph-capture-safe replacement for memset)
// ---------------------------------------------------------------------------
__global__ void cnf_zero_kernel(float* __restrict__ p, size_t n) {
    size_t i = (size_t)blockIdx.x * blockDim.x + threadIdx.x;
    size_t stride = (size_t)gridDim.x * blockDim.x;
    for (; i < n; i += stride) p[i] = 0.0f;
}

// ---------------------------------------------------------------------------
// Y[M x 64] = X[M x 64] @ W[64 x 64] + b[64], optional fused ReLU.
// One wave32 per 16-ROW BLOCK: computes the full 16x64 output slab with
// 4 accumulator fragments (64 x V_WMMA_F32_16X16X4_F32 per wave), so each
// A fragment is loaded exactly once (no 4x re-read of X per N-tile).
// Fragment layouts per CDNA5 ISA 7.12.2:
//   A 16x4 : lane%16 = M row; VGPR j + lane-half h hold K = 2h + j
//   B 4x16 : lane%16 = N col; VGPR j + half h hold K = 2h + j
//   C/D    : VGPR r, lanes0-15 -> M=r, lanes16-31 -> M=r+8; N = lane%16
// M must be a multiple of 16 (100000 and 300000 both are).
// ---------------------------------------------------------------------------
__global__ void cnf_gemm64_wmma_kernel(const float* __restrict__ X,
                                       const float* __restrict__ W,
                                       const float* __restrict__ b,
                                       float* __restrict__ Y,
                                       int M, int do_relu) {
    const int lane = threadIdx.x & 31;
    const int wave = (int)((blockIdx.x * blockDim.x + threadIdx.x) >> 5);
    const int nMblk = M >> 4;
    if (wave >= nMblk) return;                // wave-uniform: EXEC stays all-1s

    const int m0   = wave << 4;
    const int row  = lane & 15;               // M (A) / N (B,C,D) index
    const int half = lane >> 4;

    v8f acc0, acc1, acc2, acc3;               // N-tiles 0..3 (cols 0-15..48-63)
    {
        const float b0 = b[ 0 + row], b1 = b[16 + row];
        const float b2 = b[32 + row], b3 = b[48 + row];
        acc0 = (v8f){ b0,b0,b0,b0,b0,b0,b0,b0 };
        acc1 = (v8f){ b1,b1,b1,b1,b1,b1,b1,b1 };
        acc2 = (v8f){ b2,b2,b2,b2,b2,b2,b2,b2 };
        acc3 = (v8f){ b3,b3,b3,b3,b3,b3,b3,b3 };
    }

    const float* Arow = X + (size_t)(m0 + row) * D;
#pragma unroll
    for (int s = 0; s < 16; ++s) {
        const int k = 4 * s + 2 * half;
        v2f a;
        a.x = Arow[k];
        a.y = Arow[k + 1];
        const float* Wk0 = W + (size_t)k * D + row;        // row k,   col base
        const float* Wk1 = W + (size_t)(k + 1) * D + row;  // row k+1, col base
        v2f b0; b0.x = Wk0[ 0]; b0.y = Wk1[ 0];
        v2f b1; b1.x = Wk0[16]; b1.y = Wk1[16];
        v2f b2; b2.x = Wk0[32]; b2.y = Wk1[32];
        v2f b3; b3.x = Wk0[48]; b3.y = Wk1[48];
        acc0 = __builtin_amdgcn_wmma_f32_16x16x4_f32(false, a, false, b0,
                                                     (short)0, acc0, false, false);
        acc1 = __builtin_amdgcn_wmma_f32_16x16x4_f32(false, a, false, b1,
                                                     (short)0, acc1, false, false);
        acc2 = __builtin_amdgcn_wmma_f32_16x16x4_f32(false, a, false, b2,
                                                     (short)0, acc2, false, false);
        acc3 = __builtin_amdgcn_wmma_f32_16x16x4_f32(false, a, false, b3,
                                                     (short)0, acc3, false, false);
    }

    float* Yp = Y + (size_t)m0 * D;
#pragma unroll
    for (int r = 0; r < 8; ++r) {
        const size_t ro = (size_t)(r + 8 * half) * D + row;
        float v0 = acc0[r], v1 = acc1[r], v2 = acc2[r], v3 = acc3[r];
        if (do_relu) {
            v0 = fmaxf(v0, 0.0f); v1 = fmaxf(v1, 0.0f);
            v2 = fmaxf(v2, 0.0f); v3 = fmaxf(v3, 0.0f);
        }
        Yp[ro +  0] = v0;
        Yp[ro + 16] = v1;
        Yp[ro + 32] = v2;
        Yp[ro + 48] = v3;
    }
}

// ---------------------------------------------------------------------------
// Edge scatter-add: one wave per edge; each lane handles 2 features.
//   dst[didx[e]] += src[sidx[e]];  cnt[didx[e]] += 1
// Accumulators (76.8 MB / 25.6 MB) fit in the 192 MB L2 -> f32 L2 atomics.
// ---------------------------------------------------------------------------
__global__ void cnf_edge_scatter_kernel(const float* __restrict__ src,
                                        const int* __restrict__ sidx,
                                        const int* __restrict__ didx,
                                        float* __restrict__ dst,
                                        float* __restrict__ cnt,
                                        int n_edges) {
    const int lane = threadIdx.x & 31;
    int e = (int)((blockIdx.x * blockDim.x + threadIdx.x) >> 5);
    const int nw = (int)((gridDim.x * blockDim.x) >> 5);
    for (; e < n_edges; e += nw) {
        const int s = sidx[e];
        const int d = didx[e];
        const float2 v = *(const float2*)(src + (size_t)s * D + 2 * lane);
        float* dp = dst + (size_t)d * D + 2 * lane;
        __hip_atomic_fetch_add(dp,     v.x, __ATOMIC_RELAXED, __HIP_MEMORY_SCOPE_AGENT);
        __hip_atomic_fetch_add(dp + 1, v.y, __ATOMIC_RELAXED, __HIP_MEMORY_SCOPE_AGENT);
        if (lane == 0)
            __hip_atomic_fetch_add(cnt + d, 1.0f, __ATOMIC_RELAXED, __HIP_MEMORY_SCOPE_AGENT);
    }
}

// ---------------------------------------------------------------------------
// h[i, f] = (h[i, f] / max(cnt[i], 1)), optional ReLU — in place
// ---------------------------------------------------------------------------
__global__ void cnf_seg_div_kernel(float* __restrict__ h,
                                   const float* __restrict__ cnt,
                                   int n_seg, int do_relu) {
    size_t i = (size_t)blockIdx.x * blockDim.x + threadIdx.x;
    const size_t total = (size_t)n_seg * D;
    if (i >= total) return;
    const float c = fmaxf(cnt[i >> 6], 1.0f);
    float v = h[i] / c;
    if (do_relu) v = fmaxf(v, 0.0f);
    h[i] = v;
}

// ---------------------------------------------------------------------------
// kernel_launch
// Inputs (setup_inputs order):
//  0 feat_literal [NL*64] f32    4 W_l2c [64*64] f32
//  1 feat_clause  [NC*64] f32    5 b_l2c [64]    f32
//  2 lit_idx      [NE]    i32    6 W_c2l [64*64] f32
//  3 clause_idx   [NE]    i32    7 b_c2l [64]    f32
// d_out = [h_lit (NL*64) | h2_clause (NC*64)] f32
// ---------------------------------------------------------------------------
extern "C" void kernel_launch(void* const* d_in, const int* in_sizes, int n_in,
                              void* d_out, int out_size, void* d_ws, size_t ws_size,
                              hipStream_t stream) {
    const float* feat_lit    = (const float*)d_in[0];
    const float* feat_clause = (const float*)d_in[1];
    const int*   lit_idx     = (const int*)d_in[2];
    const int*   clause_idx  = (const int*)d_in[3];
    const float* W_l2c       = (const float*)d_in[4];
    const float* b_l2c       = (const float*)d_in[5];
    const float* W_c2l       = (const float*)d_in[6];
    const float* b_c2l       = (const float*)d_in[7];

    const int NL = in_sizes[0] / D;   // 100000
    const int NC = in_sizes[1] / D;   // 300000
    const int NE = in_sizes[2];       // 3000000

    float* ws = (float*)d_ws;
    // workspace layout (floats)
    float* hclause_sum = ws;                              // NC*64
    float* cnt_clause  = hclause_sum + (size_t)NC * D;    // NC
    float* cnt_lit     = cnt_clause + NC;                 // NL
    float* Wh_c2l      = cnt_lit + NL;                    // NC*64
    float* Wh_l2c      = Wh_c2l + (size_t)NC * D;         // NL*64

    float* out_hlit = (float*)d_out;                      // NL*64
    float* out_h2   = out_hlit + (size_t)NL * D;          // NC*64

    const int TPB = 256;

    // 1) zero accumulators (contiguous ws region) + h_lit output region
    {
        const size_t nz = (size_t)NC * D + NC + NL;
        cnf_zero_kernel<<<4096, TPB, 0, stream>>>(hclause_sum, nz);
        cnf_zero_kernel<<<4096, TPB, 0, stream>>>(out_hlit, (size_t)NL * D);
    }

    // 2) Wh_l2c = feat_literal @ W_l2c + b   (WMMA f32)
    {
        const int waves = NL / 16;
        const int blocks = (waves * 32 + TPB - 1) / TPB;
        cnf_gemm64_wmma_kernel<<<blocks, TPB, 0, stream>>>(feat_lit, W_l2c, b_l2c,
                                                           Wh_l2c, NL, 0);
    }

    // 3) h2_clause = relu(feat_clause @ W_l2c + b)  -> directly into d_out
    {
        const int waves = NC / 16;
        const int blocks = (waves * 32 + TPB - 1) / TPB;
        cnf_gemm64_wmma_kernel<<<blocks, TPB, 0, stream>>>(feat_clause, W_l2c, b_l2c,
                                                           out_h2, NC, 1);
    }

    // 4) scatter Wh_l2c[lit_idx] into h_clause_sum by clause_idx (+counts)
    {
        const int blocks = (NE + (TPB / 32) - 1) / (TPB / 32);
        cnf_edge_scatter_kernel<<<blocks, TPB, 0, stream>>>(Wh_l2c, lit_idx, clause_idx,
                                                            hclause_sum, cnt_clause, NE);
    }

    // 5) cembs = relu(h_clause_sum / max(cnt,1))  in place
    {
        const size_t total = (size_t)NC * D;
        const int blocks = (int)((total + TPB - 1) / TPB);
        cnf_seg_div_kernel<<<blocks, TPB, 0, stream>>>(hclause_sum, cnt_clause, NC, 1);
    }

    // 6) Wh_c2l = cembs @ W_c2l + b   (WMMA f32)
    {
        const int waves = NC / 16;
        const int blocks = (waves * 32 + TPB - 1) / TPB;
        cnf_gemm64_wmma_kernel<<<blocks, TPB, 0, stream>>>(hclause_sum, W_c2l, b_c2l,
                                                           Wh_c2l, NC, 0);
    }

    // 7) scatter Wh_c2l[clause_idx] into h_lit by lit_idx (+counts)
    {
        const int blocks = (NE + (TPB / 32) - 1) / (TPB / 32);
        cnf_edge_scatter_kernel<<<blocks, TPB, 0, stream>>>(Wh_c2l, clause_idx, lit_idx,
                                                            out_hlit, cnt_lit, NE);
    }

    // 8) h_lit /= max(cnt,1)  (no ReLU — reference returns pre-activation)
    {
        const size_t total = (size_t)NL * D;
        const int blocks = (int)((total + TPB - 1) / TPB);
        cnf_seg_div_kernel<<<blocks, TPB, 0, stream>>>(out_hlit, cnt_lit, NL, 0);
    }
}